// SpikingAttentionJAX_49718541418724
// MI455X (gfx1250) — compile-verified
//
#include <hip/hip_runtime.h>
#include <stdint.h>

#define DECAY     0.7f
#define THETA     1.0f
#define KWIN      5
#define GAIN_UP   1.5f
#define GAIN_DOWN 0.6f

// ---------------- K1: zero the histogram (d_ws is poisoned / stale) ----------------
__global__ void k_zero_counts(unsigned int* __restrict__ counts, int vocab) {
    int i = blockIdx.x * blockDim.x + threadIdx.x;
    if (i < vocab) counts[i] = 0u;
}

// ---------------- K2: token histogram, staging tokens through LDS ------------------
// Each 256-thread block handles 1024 tokens: every lane async-copies its own 16 bytes
// (4 tokens) global->LDS (gfx1250 async path, tracked by ASYNCcnt), waits, then issues
// 4 global atomicAdds. Only 8192 atomics total over a 512KB counts array -> L2-resident.
__global__ __launch_bounds__(256) void k_hist(const int* __restrict__ tokens,
                                              unsigned int* __restrict__ counts,
                                              int seq, int vocab) {
    __shared__ int toks[1024];
    const int lid  = threadIdx.x;                 // 0..255
    const int base = blockIdx.x * 1024;
    const int nfull = (seq / 1024) * 1024;

    if (base < nfull) {
#if __has_builtin(__builtin_amdgcn_global_load_async_to_lds_b128)
        typedef int v4i_t __attribute__((ext_vector_type(4)));
        typedef __attribute__((address_space(1))) v4i_t gv4_t;   // global
        typedef __attribute__((address_space(3))) v4i_t lv4_t;   // LDS
        __builtin_amdgcn_global_load_async_to_lds_b128(
            (gv4_t*)(uintptr_t)(tokens + base + lid * 4),
            (lv4_t*)(uintptr_t)(&toks[lid * 4]),
            0, 0);
#if __has_builtin(__builtin_amdgcn_s_wait_asynccnt)
        __builtin_amdgcn_s_wait_asynccnt(0);
#else
        asm volatile("s_wait_asynccnt 0" ::: "memory");
#endif
#else
        // fallback: plain b128 load + LDS store
        *(int4*)(&toks[lid * 4]) = *(const int4*)(tokens + base + lid * 4);
#endif
        __syncthreads();
#pragma unroll
        for (int j = 0; j < 4; ++j) {
            int t = toks[lid * 4 + j];
            t = (t < 0) ? 0 : ((t >= vocab) ? vocab - 1 : t);
            atomicAdd(&counts[t], 1u);
        }
    } else {
        // remainder tail (not hit for seq=8192, kept for generality)
        for (int i = base + lid; i < seq; i += 256) {
            int t = tokens[i];
            t = (t < 0) ? 0 : ((t >= vocab) ? vocab - 1 : t);
            atomicAdd(&counts[t], 1u);
        }
    }
}

// ---------------- K3: per-bin LIF recurrence -> preliminary gains ------------------
// Final membrane/spikes depend only on occurrence count; iterating the map count[i]
// times from v0[i] reproduces the reference's fp32 sequence bit-exactly.
__global__ void k_lif_gain(const unsigned int* __restrict__ counts,
                           const float* __restrict__ v0,
                           float* __restrict__ out, int vocab) {
    int i = blockIdx.x * blockDim.x + threadIdx.x;
    if (i >= vocab) return;
    unsigned int n = counts[i];
    float v = v0[i];
    unsigned int s = 0;
    for (unsigned int j = 0; j < n; ++j) {
        v = DECAY * v + 1.0f;
        if (v >= THETA) { v -= THETA; ++s; }
    }
    out[i] = (s > 0) ? GAIN_DOWN : 1.0f;
}

// ---------------- K4: global top-5 of v_final, overwrite winners -------------------
// key = ordered_float(v) << 32 | ~index : larger key == larger value, ties -> lower
// index (matches lax.top_k). Single block; v recomputed from counts (sum(count)=seq
// extra iterations across the whole block -> negligible).
__device__ __forceinline__ unsigned int ord_f32(float f) {
    unsigned int u = __float_as_uint(f);
    return (u & 0x80000000u) ? ~u : (u | 0x80000000u);
}

__global__ __launch_bounds__(256) void k_topk_mark(const unsigned int* __restrict__ counts,
                                                   const float* __restrict__ v0,
                                                   float* __restrict__ out, int vocab) {
    const int T = 256;
    __shared__ unsigned long long sm[T * KWIN];
    const int tid = threadIdx.x;

    unsigned long long lk[KWIN];
#pragma unroll
    for (int j = 0; j < KWIN; ++j) lk[j] = 0ull;   // sentinel: -inf value / invalid idx

    for (int i = tid; i < vocab; i += T) {
        unsigned int n = counts[i];
        float v = v0[i];
        for (unsigned int j = 0; j < n; ++j) {
            v = DECAY * v + 1.0f;
            if (v >= THETA) v -= THETA;
        }
        unsigned long long key = ((unsigned long long)ord_f32(v) << 32)
                               | (unsigned int)(~(unsigned int)i);
        if (key > lk[KWIN - 1]) {                  // insert into sorted-desc top-5
            int p = KWIN - 1;
            while (p > 0 && lk[p - 1] < key) { lk[p] = lk[p - 1]; --p; }
            lk[p] = key;
        }
    }
#pragma unroll
    for (int j = 0; j < KWIN; ++j) sm[tid * KWIN + j] = lk[j];
    __syncthreads();

    for (int s = T / 2; s > 0; s >>= 1) {
        if (tid < s) {
            unsigned long long m[KWIN];
            const unsigned long long* a = &sm[tid * KWIN];
            const unsigned long long* b = &sm[(tid + s) * KWIN];
            int ia = 0, ib = 0;
#pragma unroll
            for (int j = 0; j < KWIN; ++j) m[j] = (a[ia] >= b[ib]) ? a[ia++] : b[ib++];
#pragma unroll
            for (int j = 0; j < KWIN; ++j) sm[tid * KWIN + j] = m[j];
        }
        __syncthreads();
    }

    if (tid == 0) {
        for (int j = 0; j < KWIN; ++j) {
            unsigned int idx = ~(unsigned int)(sm[j] & 0xFFFFFFFFull);
            if (idx < (unsigned int)vocab) out[idx] = GAIN_UP;  // winners override 0.6
        }
    }
}

// -----------------------------------------------------------------------------------
extern "C" void kernel_launch(void* const* d_in, const int* in_sizes, int n_in,
                              void* d_out, int out_size, void* d_ws, size_t ws_size,
                              hipStream_t stream) {
    const int*   tokens = (const int*)d_in[0];
    const float* v0     = (const float*)d_in[1];
    const int    seq    = in_sizes[0];   // 8192
    const int    vocab  = in_sizes[1];   // 128000 (== out_size)
    float*       out    = (float*)d_out;
    unsigned int* counts = (unsigned int*)d_ws;  // vocab * 4 bytes of scratch

    const int vb = (vocab + 255) / 256;
    k_zero_counts<<<vb, 256, 0, stream>>>(counts, vocab);

    const int hb = (seq + 1023) / 1024;
    k_hist<<<hb, 256, 0, stream>>>(tokens, counts, seq, vocab);

    k_lif_gain<<<vb, 256, 0, stream>>>(counts, v0, out, vocab);

    k_topk_mark<<<1, 256, 0, stream>>>(counts, v0, out, vocab);
}